// GraphStructureInference_11922829214141
// MI455X (gfx1250) — compile-verified
//
#include <hip/hip_runtime.h>

typedef float v2f __attribute__((ext_vector_type(2)));
typedef float v8f __attribute__((ext_vector_type(8)));

#define Bsz   2
#define INC   256
#define Hd    128
#define Wd    128
#define HW    16384      // 128*128
#define KQC   128
#define KTOT  2304       // 256 * 9
#define SAMP  32

// K-pair-interleaved LDS tile: element (t, n) -> (t>>1)*PSTRIDE + 2*n + (t&1)
// PSTRIDE = 160 floats: 160 % 64 == 32, so the two half-wave row-groups hit
// disjoint 32-bank halves; a lane's (kl, kl+1) fragment is one aligned v2f.
#define PSTRIDE 160

// -------------------------------------------------------------------------
// Kernel 1: 3x3 SAME conv as implicit GEMM.
//   A = conv_w viewed as (128 x 2304) row-major (matches (KQ,INC,3,3) flat)
//   B = im2col(feats)  (2304 x HW), zero padded
//   x[b,m,n] = A @ B + bias
// Block: 256 thr (8 waves), tile M=128 x N=64, K staged in LDS chunks of 36.
// -------------------------------------------------------------------------
__global__ __launch_bounds__(256)
void conv3x3_gemm(const float* __restrict__ feats, const float* __restrict__ wgt,
                  const float* __restrict__ bias, float* __restrict__ xout)
{
    const int b      = blockIdx.z;
    const int n_base = blockIdx.x * 64;

    __shared__ float As[128 * 37];        // [m][t], stride 37 (odd -> no conflicts)
    __shared__ float Bs[18 * PSTRIDE];    // 36 K-rows as 18 interleaved pairs x 64 cols

    const int tid    = threadIdx.x;
    const int wave   = tid >> 5;
    const int lane   = tid & 31;
    const int hl     = lane & 15;          // M-row / N-col within fragment
    const int lanehi = lane >> 4;          // 0: K 0,1   1: K 2,3

    v8f acc[4] = {};

    for (int ch = 0; ch < 64; ++ch) {      // 64 * 36 = 2304
        const int kk0 = ch * 36;
        // stage A: 128*36 weights
        for (int i = tid; i < 128 * 36; i += 256) {
            int m = i / 36, t = i - m * 36;
            As[m * 37 + t] = wgt[m * KTOT + kk0 + t];
        }
        // stage B: im2col 36 x 64 (zero pad outside image), pair-interleaved
        for (int i = tid; i < 36 * 64; i += 256) {
            int t  = i >> 6, n = i & 63;
            int kk = kk0 + t;
            int c  = kk / 9, r = kk - c * 9;
            int dy = r / 3 - 1, dx = r - (r / 3) * 3 - 1;
            int pos = n_base + n;
            int hh  = (pos >> 7) + dy, ww = (pos & 127) + dx;
            float v = 0.f;
            if ((unsigned)hh < 128u && (unsigned)ww < 128u)
                v = feats[((b * INC + c) * Hd + hh) * Wd + ww];
            Bs[(t >> 1) * PSTRIDE + 2 * n + (t & 1)] = v;
        }
        __syncthreads();

        const float* arow = &As[(wave * 16 + hl) * 37];
        #pragma unroll
        for (int t4 = 0; t4 < 9; ++t4) {
            const int kl = t4 * 4 + lanehi * 2;
            v2f a; a.x = arow[kl]; a.y = arow[kl + 1];
            const float* brow = &Bs[(t4 * 2 + lanehi) * PSTRIDE];
            #pragma unroll
            for (int nt = 0; nt < 4; ++nt) {
                v2f bb = *(const v2f*)&brow[2 * (nt * 16 + hl)];
                acc[nt] = __builtin_amdgcn_wmma_f32_16x16x4_f32(
                    false, a, false, bb, (short)0, acc[nt], false, false);
            }
        }
        __syncthreads();
    }

    // epilogue: D layout = lanes 0-15: (M=r, N=lane); lanes 16-31: (M=r+8, N=lane-16)
    #pragma unroll
    for (int nt = 0; nt < 4; ++nt)
        #pragma unroll
        for (int r = 0; r < 8; ++r) {
            int m = wave * 16 + ((lane < 16) ? r : r + 8);
            int n = n_base + nt * 16 + hl;
            xout[(b * KQC + m) * HW + n] = acc[nt][r] + bias[m];
        }
}

// -------------------------------------------------------------------------
// Kernel 2: fused key & query projections (two 128x128 GEMMs sharing x tile).
// Writes results position-major (b, n, c) so later gathers read contiguous rows.
// -------------------------------------------------------------------------
__global__ __launch_bounds__(256)
void proj_gemm(const float* __restrict__ x,
               const float* __restrict__ kw, const float* __restrict__ kb,
               const float* __restrict__ qw, const float* __restrict__ qb,
               float* __restrict__ kout, float* __restrict__ qout)
{
    const int b      = blockIdx.z;
    const int n_base = blockIdx.x * 64;

    __shared__ float Ak[128 * 33];
    __shared__ float Aq[128 * 33];
    __shared__ float Xs[16 * PSTRIDE];    // 32 K-rows as 16 interleaved pairs

    const int tid    = threadIdx.x;
    const int wave   = tid >> 5;
    const int lane   = tid & 31;
    const int hl     = lane & 15;
    const int lanehi = lane >> 4;

    v8f acck[4] = {}, accq[4] = {};

    for (int ch = 0; ch < 4; ++ch) {       // 4 * 32 = 128 = K
        const int c0 = ch * 32;
        for (int i = tid; i < 128 * 32; i += 256) {
            int m = i >> 5, t = i & 31;
            Ak[m * 33 + t] = kw[m * KQC + c0 + t];
            Aq[m * 33 + t] = qw[m * KQC + c0 + t];
        }
        for (int i = tid; i < 32 * 64; i += 256) {
            int t = i >> 6, n = i & 63;
            Xs[(t >> 1) * PSTRIDE + 2 * n + (t & 1)] =
                x[(b * KQC + c0 + t) * HW + n_base + n];
        }
        __syncthreads();

        const float* akr = &Ak[(wave * 16 + hl) * 33];
        const float* aqr = &Aq[(wave * 16 + hl) * 33];
        #pragma unroll
        for (int t4 = 0; t4 < 8; ++t4) {
            const int kl = t4 * 4 + lanehi * 2;
            v2f ak; ak.x = akr[kl]; ak.y = akr[kl + 1];
            v2f aq; aq.x = aqr[kl]; aq.y = aqr[kl + 1];
            const float* xrow = &Xs[(t4 * 2 + lanehi) * PSTRIDE];
            #pragma unroll
            for (int nt = 0; nt < 4; ++nt) {
                v2f xb = *(const v2f*)&xrow[2 * (nt * 16 + hl)];
                acck[nt] = __builtin_amdgcn_wmma_f32_16x16x4_f32(
                    false, ak, false, xb, (short)0, acck[nt], false, false);
                accq[nt] = __builtin_amdgcn_wmma_f32_16x16x4_f32(
                    false, aq, false, xb, (short)0, accq[nt], false, false);
            }
        }
        __syncthreads();
    }

    #pragma unroll
    for (int nt = 0; nt < 4; ++nt)
        #pragma unroll
        for (int r = 0; r < 8; ++r) {
            int m = wave * 16 + ((lane < 16) ? r : r + 8);
            int n = n_base + nt * 16 + hl;
            kout[(b * HW + n) * KQC + m] = acck[nt][r] + kb[m];
            qout[(b * HW + n) * KQC + m] = accq[nt][r] + qb[m];
        }
}

// -------------------------------------------------------------------------
// Kernel 3: s x s average pool, (b,n,c) -> (b,n_out,c). Threads = channels.
// -------------------------------------------------------------------------
__global__ __launch_bounds__(128)
void pool_kernel(const float* __restrict__ src, float* __restrict__ dst,
                 int s, int w_out)
{
    const int b = blockIdx.z;
    const int n = blockIdx.x;                // output position
    const int c = threadIdx.x;
    const int h = n / w_out, w = n - h * w_out;
    float sum = 0.f;
    for (int i = 0; i < s; ++i)
        for (int j = 0; j < s; ++j) {
            int pos = (h * s + i) * Wd + (w * s + j);
            sum += src[((size_t)b * HW + pos) * KQC + c];
        }
    dst[((size_t)b * (w_out * w_out) + n) * KQC + c] = sum / (float)(s * s);
}

// -------------------------------------------------------------------------
// Kernel 4: gathered attention logits. One wave per (b,n), lane = sample.
// logits[b,n,s] = dot(q[b,n,:], k[b, inds[b,n,s], :]) * 128^-0.5
// -------------------------------------------------------------------------
__global__ __launch_bounds__(256)
void logits_kernel(const float* __restrict__ q, const float* __restrict__ k,
                   const int* __restrict__ inds, float* __restrict__ out,
                   int n_level, int out_off)
{
    const int gw   = (int)((blockIdx.x * blockDim.x + threadIdx.x) >> 5);
    const int lane = threadIdx.x & 31;
    if (gw >= Bsz * n_level) return;         // whole wave exits together
    const int b = gw / n_level;
    const int n = gw - b * n_level;

    const int ind = inds[((size_t)(b * n_level + n)) * SAMP + lane];
    const float4* qr = (const float4*)&q[((size_t)b * n_level + n) * KQC];
    const float4* kr = (const float4*)&k[((size_t)b * n_level + ind) * KQC];

    float acc = 0.f;
    #pragma unroll
    for (int i = 0; i < KQC / 4; ++i) {
        float4 qa = qr[i], ka = kr[i];
        acc += qa.x * ka.x + qa.y * ka.y + qa.z * ka.z + qa.w * ka.w;
    }
    out[((size_t)b * 21504 + out_off + n) * SAMP + lane] =
        acc * 0.08838834764831845f;          // 128^-0.5
}

// -------------------------------------------------------------------------
extern "C" void kernel_launch(void* const* d_in, const int* in_sizes, int n_in,
                              void* d_out, int out_size, void* d_ws, size_t ws_size,
                              hipStream_t stream)
{
    const float* feats  = (const float*)d_in[0];
    const float* conv_w = (const float*)d_in[1];
    const float* conv_b = (const float*)d_in[2];
    const float* key_w  = (const float*)d_in[3];
    const float* key_b  = (const float*)d_in[4];
    const float* qry_w  = (const float*)d_in[5];
    const float* qry_b  = (const float*)d_in[6];
    const int*   inds1  = (const int*)d_in[7];   // stride 1, n=16384
    const int*   inds2  = (const int*)d_in[8];   // stride 2, n=4096
    const int*   inds4  = (const int*)d_in[9];   // stride 4, n=1024

    float* ws = (float*)d_ws;
    float* x  = ws;                       // (b,c,n)       4,194,304
    float* k1 = ws + 4194304;             // (b,n,c)       4,194,304
    float* q1 = ws + 8388608;             //               4,194,304
    float* k2 = ws + 12582912;            // (b,4096,c)    1,048,576
    float* q2 = ws + 13631488;
    float* k4 = ws + 14680064;            // (b,1024,c)      262,144
    float* q4 = ws + 14942208;
    float* out = (float*)d_out;

    // 1) conv 3x3 SAME -> x
    conv3x3_gemm<<<dim3(HW / 64, 1, Bsz), 256, 0, stream>>>(feats, conv_w, conv_b, x);

    // 2) key/query projections -> (b,n,c)
    proj_gemm<<<dim3(HW / 64, 1, Bsz), 256, 0, stream>>>(x, key_w, key_b,
                                                         qry_w, qry_b, k1, q1);

    // 3) avg pools for strides 2 and 4
    pool_kernel<<<dim3(64 * 64, 1, Bsz), 128, 0, stream>>>(k1, k2, 2, 64);
    pool_kernel<<<dim3(64 * 64, 1, Bsz), 128, 0, stream>>>(q1, q2, 2, 64);
    pool_kernel<<<dim3(32 * 32, 1, Bsz), 128, 0, stream>>>(k1, k4, 4, 32);
    pool_kernel<<<dim3(32 * 32, 1, Bsz), 128, 0, stream>>>(q1, q4, 4, 32);

    // 4) gathered logits, concatenated along n: [0,16384) [16384,20480) [20480,21504)
    logits_kernel<<<dim3((Bsz * 16384 + 7) / 8, 1, 1), 256, 0, stream>>>(
        q1, k1, inds1, out, 16384, 0);
    logits_kernel<<<dim3((Bsz * 4096 + 7) / 8, 1, 1), 256, 0, stream>>>(
        q2, k2, inds2, out, 4096, 16384);
    logits_kernel<<<dim3((Bsz * 1024 + 7) / 8, 1, 1), 256, 0, stream>>>(
        q4, k4, inds4, out, 1024, 20480);
}